// FinalTwoBranchDH_SFNN_86474871538217
// MI455X (gfx1250) — compile-verified
//
#include <hip/hip_runtime.h>
#include <cstddef>
#include <cstdint>

// Problem constants (match reference setup_inputs)
#define BB     256
#define TT     4096
#define II     40
#define HH     64
#define HALF   20
#define CHUNK  16
#define NCHUNK (TT / CHUNK)

#define XS_STRIDE 41                         // pad 40 -> 41: bank-conflict-free A-frag reads
#define XS_FLOATS (CHUNK * XS_STRIDE)        // 656 floats per buffer
#define XS_BYTES  (XS_FLOATS * 4)            // 2624 B per buffer
#define DS_STRIDE 68                         // pad 64 -> 68: conflict-free writer+reader

typedef float v2f __attribute__((ext_vector_type(2)));
typedef float v8f __attribute__((ext_vector_type(8)));

__device__ __forceinline__ float sigmoidf_fast(float v) {
    return 1.0f / (1.0f + __expf(-v));
}

__device__ __forceinline__ void wait_asynccnt0() {
#if defined(__has_builtin) && __has_builtin(__builtin_amdgcn_s_wait_asynccnt)
    __builtin_amdgcn_s_wait_asynccnt(0);
#else
    asm volatile("s_wait_asynccnt 0x0" ::: "memory");
#endif
}

__global__ __launch_bounds__(128) void sfnn_fused_kernel(
    const float* __restrict__ x,     // [B,T,I]
    const float* __restrict__ W1,    // [H,HALF]
    const float* __restrict__ b1,    // [H]
    const float* __restrict__ W2,    // [H,HALF]
    const float* __restrict__ b2,    // [H]
    const float* __restrict__ Wo,    // [1,H]
    const float* __restrict__ bo,    // [1]
    const float* __restrict__ tau_m, // [H]
    const float* __restrict__ tau_n1,// [H]
    const float* __restrict__ tau_n2,// [H]
    float* __restrict__ out)         // [B,T,1]
{
    __shared__ float xs [2 * XS_FLOATS];      // double-buffered staged x chunks
    __shared__ float d1s[CHUNK * DS_STRIDE];  // d1_in tile [16 t][64 h]
    __shared__ float d2s[CHUNK * DS_STRIDE];  // d2_in tile
    __shared__ float mo [CHUNK * DS_STRIDE];  // Wo[h]*mem  [16 t][64 h]

    const int tid  = threadIdx.x;
    const int b    = blockIdx.x;
    const int lane = tid & 31;
    const int wave = tid >> 5;

    // WMMA role of this lane: B/C/D column h, A row m, K half select
    const int hl    = (wave << 4) + (lane & 15);
    const int khalf = (lane >> 4) << 1;     // 0 for lanes 0-15, 2 for lanes 16-31
    const int m     = lane & 15;            // A-matrix row (= local t)
    const int trow  = (lane >> 4) << 3;     // D-matrix: M = r + 8*(lane>=16)
    const int mrow  = m * XS_STRIDE;        // LDS row base (floats) of this lane's t-row

    // ---- Precompute per-thread async-copy addresses (chunk-invariant) ----
    // Chunk = 640 contiguous floats in global; thread owns i = tid + 128*j, j=0..4.
    // LDS destination uses the padded [t][41] layout.
    uint32_t ldsA[5];
#pragma unroll
    for (int j = 0; j < 5; ++j) {
        const int i = tid + 128 * j;
        ldsA[j] = (uint32_t)(uintptr_t)&xs[(i / II) * XS_STRIDE + (i % II)];
    }
    const uint64_t xgbase = (uint64_t)(uintptr_t)(x + (size_t)b * TT * II);
    const uint32_t voff0  = (uint32_t)tid * 4u;

    // ---- Preload B-operand weight fragments into registers (time-invariant) ----
    v2f w1f[5], w2f[5];
#pragma unroll
    for (int ks = 0; ks < 5; ++ks) {
        const int k = ks * 4 + khalf;
        w1f[ks].x = W1[hl * HALF + k];
        w1f[ks].y = W1[hl * HALF + k + 1];
        w2f[ks].x = W2[hl * HALF + k];
        w2f[ks].y = W2[hl * HALF + k + 1];
    }
    const float bias1 = b1[hl];
    const float bias2 = b2[hl];

    // ---- Scan-thread state (threads 0..63 own channel h = tid) ----
    float d1 = 0.0f, d2 = 0.0f, mem = 0.0f;
    float al = 0.0f, be1 = 0.0f, be2 = 0.0f, woh = 0.0f;
    if (tid < HH) {
        al  = sigmoidf_fast(tau_m [tid]);
        be1 = sigmoidf_fast(tau_n1[tid]);
        be2 = sigmoidf_fast(tau_n2[tid]);
        woh = Wo[tid];
    }
    const float bov = bo[0];

    float* __restrict__ outb = out + (size_t)b * TT;

    // Issue a chunk's async global->LDS fill (5 x b32 per thread, ASYNCcnt-tracked).
    auto issue_chunk = [&](int c) {
        const uint32_t bufofs = (uint32_t)(c & 1) * (uint32_t)XS_BYTES;
        const uint32_t vbase  = voff0 + (uint32_t)c * (uint32_t)(CHUNK * II * 4);
#pragma unroll
        for (int j = 0; j < 5; ++j) {
            asm volatile("global_load_async_to_lds_b32 %0, %1, %2"
                         :
                         : "v"(ldsA[j] + bufofs), "v"(vbase + (uint32_t)(j * 512)),
                           "s"(xgbase)
                         : "memory");
        }
    };

    // ---- Software pipeline: prefetch chunk 0, then overlap fill(c+1) with compute(c)
    issue_chunk(0);

    for (int c = 0; c < NCHUNK; ++c) {
        wait_asynccnt0();        // our fill of buf[c&1] complete
        __syncthreads();         // everyone's fill complete -> buffer readable

        if (c + 1 < NCHUNK) issue_chunk(c + 1);   // fill other buffer in background

        const float* __restrict__ xb = &xs[(c & 1) * XS_FLOATS];

        // ---- WMMA: D[t][h] = bias + sum_k x[t][k] * W[h][k]  (two K=20 GEMMs)
        v8f c1, c2;
#pragma unroll
        for (int e = 0; e < 8; ++e) { c1[e] = bias1; c2[e] = bias2; }

#pragma unroll
        for (int ks = 0; ks < 5; ++ks) {
            const int kb = ks * 4 + khalf;
            v2f a1, a2;
            a1.x = xb[mrow + kb];
            a1.y = xb[mrow + kb + 1];
            a2.x = xb[mrow + HALF + kb];
            a2.y = xb[mrow + HALF + kb + 1];
            c1 = __builtin_amdgcn_wmma_f32_16x16x4_f32(
                    false, a1, false, w1f[ks], (short)0, c1, false, false);
            c2 = __builtin_amdgcn_wmma_f32_16x16x4_f32(
                    false, a2, false, w2f[ks], (short)0, c2, false, false);
        }

        // D layout: VGPR r, this lane -> (t = trow + r, h = hl)
#pragma unroll
        for (int r = 0; r < 8; ++r) {
            d1s[(trow + r) * DS_STRIDE + hl] = c1[r];
            d2s[(trow + r) * DS_STRIDE + hl] = c2[r];
        }
        __syncthreads();

        // ---- Sequential leaky-integrator scan over the 16 timesteps
        if (tid < HH) {
#pragma unroll
            for (int t = 0; t < CHUNK; ++t) {
                d1  = be1 * d1 + (1.0f - be1) * d1s[t * DS_STRIDE + tid];
                d2  = be2 * d2 + (1.0f - be2) * d2s[t * DS_STRIDE + tid];
                mem = al  * mem + (1.0f - al) * (d1 + d2);
                mo[t * DS_STRIDE + tid] = woh * mem;
            }
        }
        __syncthreads();

        // ---- Output projection (reduce over h) + sigmoid
        if (tid < CHUNK) {
            float s = bov;
#pragma unroll
            for (int h = 0; h < HH; ++h) s += mo[tid * DS_STRIDE + h];
            outb[c * CHUNK + tid] = sigmoidf_fast(s);
        }
        __syncthreads();
    }
}

extern "C" void kernel_launch(void* const* d_in, const int* in_sizes, int n_in,
                              void* d_out, int out_size, void* d_ws, size_t ws_size,
                              hipStream_t stream) {
    (void)in_sizes; (void)n_in; (void)out_size; (void)d_ws; (void)ws_size;
    const float* x      = (const float*)d_in[0];
    const float* W1     = (const float*)d_in[1];
    const float* b1     = (const float*)d_in[2];
    const float* W2     = (const float*)d_in[3];
    const float* b2     = (const float*)d_in[4];
    const float* Wo     = (const float*)d_in[5];
    const float* bo     = (const float*)d_in[6];
    const float* tau_m  = (const float*)d_in[7];
    const float* tau_n1 = (const float*)d_in[8];
    const float* tau_n2 = (const float*)d_in[9];
    float* out = (float*)d_out;

    sfnn_fused_kernel<<<BB, 128, 0, stream>>>(
        x, W1, b1, W2, b2, Wo, bo, tau_m, tau_n1, tau_n2, out);
}